// ReversePolicy_52080773431693
// MI455X (gfx1250) — compile-verified
//
#include <hip/hip_runtime.h>
#include <hip/hip_bf16.h>

typedef __attribute__((ext_vector_type(2))) float v2f;
typedef __attribute__((ext_vector_type(8))) float v8f;

// Problem constants (from reference): B=32, N=128, D=128, E=32, M=8192
#define PB 32
#define PN 128
#define PD 128
#define PE 32
#define PM 8192
#define PK (2 * PD + PE)   // 288 feature length
#define OUTW (PM + 1)      // 8193 logits per batch

// ---------------------------------------------------------------------------
// Edit-logit kernel.
// Block = 256 threads (8 waves). Each wave computes one 16-edit tile for one
// batch with V_WMMA_F32_16X16X4_F32: D(16x16) += A(16x4) * B(4x16).
//   A: 16 gathered feature rows, K-slice of 4 (float2 per lane, streamed)
//   B: W_edit rows zero-padded to 16 columns, staged in LDS
// grid = (M/(16*8), B) = (64, 32)
// ---------------------------------------------------------------------------
__global__ __launch_bounds__(256) void edit_logits_kernel(
    const float* __restrict__ h_nodes,   // [B,N,D]
    const float* __restrict__ h_edges,   // [B,N,N,E]
    const float* __restrict__ W_edit,    // [288,4]
    const float* __restrict__ b_edit,    // [4]
    const int*   __restrict__ i_idx,     // [M]
    const int*   __restrict__ j_idx,     // [M]
    const int*   __restrict__ b_idx,     // [M]
    const unsigned char* __restrict__ feas,  // [B,M] bool
    float* __restrict__ out)             // [B, M+1]
{
    __shared__ float Wp[PK * 16];        // W_edit padded to 288x16 (18 KB)
    __shared__ float tl[8][16 * 4];      // per-wave 16x4 logit scratch (2 KB)

    const int tid = threadIdx.x;
    const int b  = blockIdx.y;

    // Stage zero-padded weights: Wp[k*16+n] = (n<4) ? W_edit[k*4+n] : 0
    for (int t = tid; t < PK * 16; t += 256) {
        const int k = t >> 4;
        const int n = t & 15;
        Wp[t] = (n < 4) ? W_edit[k * 4 + n] : 0.0f;
    }
    __syncthreads();

    const int wave = tid >> 5;
    const int lane = tid & 31;
    const int n    = lane & 15;          // A: row (edit), B: column (channel)
    const int h2   = (lane >> 4) << 1;   // K sub-offset: halves hold K {0,1} / {2,3}
    const int m_base = (blockIdx.x * 8 + wave) * 16;
    const int m = m_base + n;

    const int i = i_idx[m];
    const int j = j_idx[m];
    const float* pi = h_nodes + ((size_t)b * PN + i) * PD;
    const float* pj = h_nodes + ((size_t)b * PN + j) * PD;
    const float* pe = h_edges + (((size_t)b * PN + (size_t)i) * PN + j) * PE;

    v8f acc = {};

    // Segment 1: node_i features, K = 0..127
    #pragma unroll 4
    for (int kb = 0; kb < PD; kb += 4) {
        const int k = kb + h2;
        v2f a = *(const v2f*)(pi + k);
        v2f w;
        w.x = Wp[k * 16 + n];
        w.y = Wp[(k + 1) * 16 + n];
        acc = __builtin_amdgcn_wmma_f32_16x16x4_f32(
            false, a, false, w, (short)0, acc, false, false);
    }
    // Segment 2: node_j features, K = 128..255
    #pragma unroll 4
    for (int kb = 0; kb < PD; kb += 4) {
        const int k = kb + h2;
        v2f a = *(const v2f*)(pj + k);
        const int kw = PD + k;
        v2f w;
        w.x = Wp[kw * 16 + n];
        w.y = Wp[(kw + 1) * 16 + n];
        acc = __builtin_amdgcn_wmma_f32_16x16x4_f32(
            false, a, false, w, (short)0, acc, false, false);
    }
    // Segment 3: edge features, K = 256..287
    #pragma unroll
    for (int kb = 0; kb < PE; kb += 4) {
        const int k = kb + h2;
        v2f a = *(const v2f*)(pe + k);
        const int kw = 2 * PD + k;
        v2f w;
        w.x = Wp[kw * 16 + n];
        w.y = Wp[(kw + 1) * 16 + n];
        acc = __builtin_amdgcn_wmma_f32_16x16x4_f32(
            false, a, false, w, (short)0, acc, false, false);
    }

    // D layout: VGPR v, lanes 0-15 -> (M=v, N=lane), lanes 16-31 -> (M=v+8, N=lane-16).
    // Only channels N<4 are real; spill 16x4 useful logits through LDS.
    if (n < 4) {
        const int half = lane >> 4;
        #pragma unroll
        for (int v = 0; v < 8; ++v) {
            const int r = v + half * 8;
            tl[wave][r * 4 + n] = acc[v];
        }
    }
    __syncthreads();

    // Channel select + bias + feasibility mask, one lane per edit row.
    if (lane < 16) {
        const int r  = lane;
        const int mm = m_base + r;
        const int ch = b_idx[mm];
        const float val = tl[wave][r * 4 + ch] + b_edit[ch];
        const bool ok = feas[(size_t)b * PM + mm] != 0;
        out[(size_t)b * OUTW + mm] = ok ? val : -__builtin_inff();
    }
}

// ---------------------------------------------------------------------------
// STOP-logit kernel: one block per batch.
// stop[b] = mean_n(h_nodes[b]) . W_stop + b_stop, masked by stop_feas.
// ---------------------------------------------------------------------------
__global__ __launch_bounds__(256) void stop_logit_kernel(
    const float* __restrict__ h_nodes,       // [B,N,D]
    const float* __restrict__ W_stop,        // [D,1]
    const float* __restrict__ b_stop,        // [1]
    const unsigned char* __restrict__ stop_feas, // [B] bool
    float* __restrict__ out)                 // [B, M+1]
{
    const int b   = blockIdx.x;
    const int tid = threadIdx.x;
    const float* base = h_nodes + (size_t)b * PN * PD;

    float p = 0.0f;
    for (int idx = tid; idx < PN * PD; idx += 256)
        p += base[idx] * W_stop[idx & (PD - 1)];

    __shared__ float red[256];
    red[tid] = p;
    __syncthreads();
    #pragma unroll
    for (int s = 128; s > 0; s >>= 1) {
        if (tid < s) red[tid] += red[tid + s];
        __syncthreads();
    }
    if (tid == 0) {
        const float v = red[0] * (1.0f / (float)PN) + b_stop[0];
        out[(size_t)b * OUTW + PM] = (stop_feas[b] != 0) ? v : -__builtin_inff();
    }
}

extern "C" void kernel_launch(void* const* d_in, const int* in_sizes, int n_in,
                              void* d_out, int out_size, void* d_ws, size_t ws_size,
                              hipStream_t stream) {
    const float* h_nodes = (const float*)d_in[0];
    const float* h_edges = (const float*)d_in[1];
    const float* W_edit  = (const float*)d_in[2];
    const float* b_edit  = (const float*)d_in[3];
    const float* W_stop  = (const float*)d_in[4];
    const float* b_stop  = (const float*)d_in[5];
    const int*   i_idx   = (const int*)d_in[6];
    const int*   j_idx   = (const int*)d_in[7];
    const int*   b_idx   = (const int*)d_in[8];
    const unsigned char* feas      = (const unsigned char*)d_in[9];
    const unsigned char* stop_feas = (const unsigned char*)d_in[10];
    float* out = (float*)d_out;

    // 512 tiles of 16 edits per batch, 8 tiles (waves) per block.
    dim3 grid(PM / (16 * 8), PB);
    edit_logits_kernel<<<grid, 256, 0, stream>>>(
        h_nodes, h_edges, W_edit, b_edit, i_idx, j_idx, b_idx, feas, out);

    stop_logit_kernel<<<PB, 256, 0, stream>>>(
        h_nodes, W_stop, b_stop, stop_feas, out);
}